// GAT_5566277616139
// MI455X (gfx1250) — compile-verified
//
#include <hip/hip_runtime.h>
#include <stdint.h>

#define NEG_SLOPE 0.2f

typedef __attribute__((ext_vector_type(2))) float v2f;
typedef __attribute__((ext_vector_type(8))) float v8f;

// ---------------------------------------------------------------------------
// GEMM via native fp32 WMMA: H[M,Fout] = X[M,K] * W[K,Fout]
// One wave per 16-row tile; NT = Fout/16 accumulator tiles held in registers.
// A fragment (16x4 f32): lane l holds M=l&15, K = k0 + 2*(l>>4) + {0,1}
// B fragment (4x16 f32): lane l holds N=l&15, K = k0 + 2*(l>>4) + {0,1}
// C/D (16x16 f32): vgpr r, lane l -> M = r + 8*(l>>4), N = l&15
// ---------------------------------------------------------------------------
template <int NT>
__global__ __launch_bounds__(256)
void gat_gemm_wmma_f32(const float* __restrict__ X, const float* __restrict__ W,
                       float* __restrict__ H, int M, int K, int Fout) {
  const int wave = (int)(blockIdx.x * (blockDim.x >> 5) + (threadIdx.x >> 5));
  const int lane = (int)(threadIdx.x & 31);
  const int row0 = wave * 16;
  if (row0 >= M) return;

  const int m  = lane & 15;
  const int ks = lane >> 4;   // 0 or 1

  v8f zero = {};
  v8f acc[NT];
#pragma unroll
  for (int c = 0; c < NT; ++c) acc[c] = zero;

  const float* xrow = X + (size_t)(row0 + m) * K;
  for (int k0 = 0; k0 < K; k0 += 4) {
    const int ka = k0 + 2 * ks;
    v2f a;
    a.x = xrow[ka];
    a.y = xrow[ka + 1];
#pragma unroll
    for (int c = 0; c < NT; ++c) {
      const int n = c * 16 + m;
      v2f b;
      b.x = W[(size_t)ka * Fout + n];
      b.y = W[(size_t)(ka + 1) * Fout + n];
      acc[c] = __builtin_amdgcn_wmma_f32_16x16x4_f32(
          /*neg_a=*/false, a, /*neg_b=*/false, b,
          /*c_mod=*/(short)0, acc[c], /*reuse_a=*/false, /*reuse_b=*/false);
    }
  }

#pragma unroll
  for (int c = 0; c < NT; ++c) {
    float* hr = H + (size_t)(row0 + 8 * ks) * Fout + c * 16 + m;
#pragma unroll
    for (int r = 0; r < 8; ++r) hr[(size_t)r * Fout] = acc[c][r];
  }
}

// ---------------------------------------------------------------------------
// Per-node attention logits: As[n] = h[n].a_src, Ad[n] = h[n].a_dst
// One wave per node, cross-lane reduction via shuffles.
// ---------------------------------------------------------------------------
__global__ __launch_bounds__(256)
void gat_node_alphas(const float* __restrict__ H, const float* __restrict__ asrc,
                     const float* __restrict__ adst, float* __restrict__ As,
                     float* __restrict__ Ad, int N, int F) {
  const int node = (int)(blockIdx.x * (blockDim.x >> 5) + (threadIdx.x >> 5));
  const int lane = (int)(threadIdx.x & 31);
  if (node >= N) return;
  const float* hr = H + (size_t)node * F;
  float ss = 0.f, sd = 0.f;
  for (int f = lane; f < F; f += 32) {
    float h = hr[f];
    ss += h * asrc[f];
    sd += h * adst[f];
  }
#pragma unroll
  for (int o = 16; o > 0; o >>= 1) {
    ss += __shfl_xor(ss, o, 32);
    sd += __shfl_xor(sd, o, 32);
  }
  if (lane == 0) { As[node] = ss; Ad[node] = sd; }
}

__global__ void gat_node_init(unsigned* __restrict__ mkey, float* __restrict__ ssum, int N) {
  int i = (int)(blockIdx.x * blockDim.x + threadIdx.x);
  if (i < N) { mkey[i] = 0u; ssum[i] = 0.f; }
}

__global__ void gat_fzero(float* __restrict__ p, int n) {
  int i = (int)(blockIdx.x * blockDim.x + threadIdx.x);
  if (i < n) p[i] = 0.f;
}

__device__ __forceinline__ void edge_sd(const long long* __restrict__ ei, int E,
                                        int t, int& s, int& d) {
  if (t < E) {
    s = (int)ei[t];
    d = (int)ei[(size_t)E + t];
  } else {
    s = d = t - E;  // self loop
  }
}

// pass 1: e = leakyrelu(As[src] + Ad[dst]); segment-max over dst via
// monotone uint encoding + atomicMax (L2 atomic).
__global__ __launch_bounds__(256)
void gat_edge_logits_max(const long long* __restrict__ ei, int E, int Etot,
                         const float* __restrict__ As, const float* __restrict__ Ad,
                         float* __restrict__ ebuf, unsigned* __restrict__ mkey) {
  int t = (int)(blockIdx.x * blockDim.x + threadIdx.x);
  if (t >= Etot) return;
  int s, d;
  edge_sd(ei, E, t, s, d);
  float e = As[s] + Ad[d];
  e = e > 0.f ? e : NEG_SLOPE * e;
  ebuf[t] = e;
  unsigned u = __float_as_uint(e);
  unsigned key = (u & 0x80000000u) ? ~u : (u | 0x80000000u);
  atomicMax(&mkey[d], key);
}

// pass 2: w = exp(e - m[dst]); segment-sum over dst.
__global__ __launch_bounds__(256)
void gat_edge_exp_sum(const long long* __restrict__ ei, int E, int Etot,
                      const unsigned* __restrict__ mkey, float* __restrict__ ebuf,
                      float* __restrict__ ssum) {
  int t = (int)(blockIdx.x * blockDim.x + threadIdx.x);
  if (t >= Etot) return;
  int s, d;
  edge_sd(ei, E, t, s, d);
  unsigned key = mkey[d];
  unsigned u = (key & 0x80000000u) ? (key & 0x7FFFFFFFu) : ~key;
  float m = __uint_as_float(u);
  float w = __expf(ebuf[t] - m);
  ebuf[t] = w;
  atomicAdd(&ssum[d], w);
}

// pass 3: out[dst] += h[src] * (w / s[dst]); one wave per edge, h gather is
// L2-resident, scatter via global_atomic_add_f32.
__global__ __launch_bounds__(256)
void gat_edge_aggregate(const long long* __restrict__ ei, int E, int Etot,
                        const float* __restrict__ H, const float* __restrict__ ebuf,
                        const float* __restrict__ ssum, float* __restrict__ out, int F) {
  const int gw = (int)(blockIdx.x * (blockDim.x >> 5) + (threadIdx.x >> 5));
  const int lane = (int)(threadIdx.x & 31);
  if (gw >= Etot) return;
  int s, d;
  edge_sd(ei, E, gw, s, d);
  const float coeff = ebuf[gw] / ssum[d];
  const float* hs = H + (size_t)s * F;
  float* od = out + (size_t)d * F;
  for (int f = lane; f < F; f += 32) atomicAdd(&od[f], hs[f] * coeff);
}

__global__ void gat_bias_act(float* __restrict__ out, const float* __restrict__ b,
                             int total, int Fmask, int relu) {
  int i = (int)(blockIdx.x * blockDim.x + threadIdx.x);
  if (i >= total) return;
  float v = out[i] + b[i & Fmask];
  out[i] = (relu && v < 0.f) ? 0.f : v;
}

// ---------------------------------------------------------------------------
// Orchestration
// ---------------------------------------------------------------------------
static inline int ceil_div(int a, int b) { return (a + b - 1) / b; }

extern "C" void kernel_launch(void* const* d_in, const int* in_sizes, int n_in,
                              void* d_out, int out_size, void* d_ws, size_t ws_size,
                              hipStream_t stream) {
  const float*     x    = (const float*)d_in[0];
  const long long* ei   = (const long long*)d_in[1];  // int64 [2,E]
  const float*     W1   = (const float*)d_in[2];
  const float*     a1s  = (const float*)d_in[3];
  const float*     a1d  = (const float*)d_in[4];
  const float*     b1   = (const float*)d_in[5];
  const float*     W2   = (const float*)d_in[6];
  const float*     a2s  = (const float*)d_in[7];
  const float*     a2d  = (const float*)d_in[8];
  const float*     b2   = (const float*)d_in[9];
  float*           out  = (float*)d_out;

  const int Fh   = in_sizes[3];            // 128
  const int Fout = in_sizes[7];            // 64
  const int Fin  = in_sizes[2] / Fh;       // 128
  const int N    = in_sizes[0] / Fin;      // 50000
  const int E    = in_sizes[1] / 2;        // 640000
  const int Etot = E + N;                  // + self loops

  // workspace layout (floats)
  float* ws   = (float*)d_ws;
  float* h    = ws;                               // N*Fh (reused as h2: N*Fout)
  float* out1 = h + (size_t)N * Fh;               // N*Fh
  float* As   = out1 + (size_t)N * Fh;            // N
  float* Ad   = As + N;                           // N
  unsigned* mkey = (unsigned*)(Ad + N);           // N
  float* ssum = (float*)(mkey + N);               // N
  float* ebuf = ssum + N;                         // Etot

  const int TB = 256;
  const int waves_per_blk = TB / 32;
  const int gemm_blks  = ceil_div(ceil_div(N, 16), waves_per_blk);
  const int node_wblks = ceil_div(N, waves_per_blk);
  const int node_tblks = ceil_div(N, TB);
  const int edge_tblks = ceil_div(Etot, TB);
  const int edge_wblks = ceil_div(Etot, waves_per_blk);

  // ---------------- Layer 1 (Fin -> Fh) ----------------
  gat_gemm_wmma_f32<8><<<gemm_blks, TB, 0, stream>>>(x, W1, h, N, Fin, Fh);
  gat_node_alphas<<<node_wblks, TB, 0, stream>>>(h, a1s, a1d, As, Ad, N, Fh);
  gat_node_init<<<node_tblks, TB, 0, stream>>>(mkey, ssum, N);
  gat_fzero<<<ceil_div(N * Fh, TB), TB, 0, stream>>>(out1, N * Fh);
  gat_edge_logits_max<<<edge_tblks, TB, 0, stream>>>(ei, E, Etot, As, Ad, ebuf, mkey);
  gat_edge_exp_sum<<<edge_tblks, TB, 0, stream>>>(ei, E, Etot, mkey, ebuf, ssum);
  gat_edge_aggregate<<<edge_wblks, TB, 0, stream>>>(ei, E, Etot, h, ebuf, ssum, out1, Fh);
  gat_bias_act<<<ceil_div(N * Fh, TB), TB, 0, stream>>>(out1, b1, N * Fh, Fh - 1, 1);

  // ---------------- Layer 2 (Fh -> Fout) ----------------
  float* h2 = h;  // reuse (N*Fout <= N*Fh)
  gat_gemm_wmma_f32<4><<<gemm_blks, TB, 0, stream>>>(out1, W2, h2, N, Fh, Fout);
  gat_node_alphas<<<node_wblks, TB, 0, stream>>>(h2, a2s, a2d, As, Ad, N, Fout);
  gat_node_init<<<node_tblks, TB, 0, stream>>>(mkey, ssum, N);
  gat_fzero<<<ceil_div(N * Fout, TB), TB, 0, stream>>>(out, N * Fout);
  gat_edge_logits_max<<<edge_tblks, TB, 0, stream>>>(ei, E, Etot, As, Ad, ebuf, mkey);
  gat_edge_exp_sum<<<edge_tblks, TB, 0, stream>>>(ei, E, Etot, mkey, ebuf, ssum);
  gat_edge_aggregate<<<edge_wblks, TB, 0, stream>>>(ei, E, Etot, h2, ebuf, ssum, out, Fout);
  gat_bias_act<<<ceil_div(N * Fout, TB), TB, 0, stream>>>(out, b2, N * Fout, Fout - 1, 0);

  (void)n_in; (void)out_size; (void)ws_size;
}